// Dilation1D_79078937854427
// MI455X (gfx1250) — compile-verified
//
#include <hip/hip_runtime.h>

// 1-D grayscale dilation (max-plus conv), K=11 parabolic SE, fp32.
// Bandwidth-bound: 256 MB traffic -> ~11 us floor at 23.3 TB/s.
// CDNA5 path: async global->LDS b128 staging (ASYNCcnt) + s_wait_asynccnt,
// v_max3-friendly compute, b128 LDS reads and global stores.

#define KSIZE    11
#define KHALF    5
#define TILE     2048      // outputs per block
#define HALO     8         // staged halo (>= KHALF, keeps 16B alignment)
#define NTHREADS 256       // 8 wave32s
#define VPT      8         // outputs per thread

__global__ __launch_bounds__(NTHREADS)
void dilation1d_gfx1250_kernel(const float* __restrict__ x,
                               const float* __restrict__ scale_p,
                               float* __restrict__ out, int n) {
    __shared__ __align__(16) float smem[TILE + 2 * HALO];   // 2064 floats

    const int t      = threadIdx.x;
    const int base   = (int)blockIdx.x * TILE;
    const int gstart = base - HALO;                 // staged global start
    const int stage  = TILE + 2 * HALO;             // 2064 floats
    const int nvec   = stage / 4;                   // 516 float4s

    // Runtime structuring-element scale (uniform scalar load).
    const float s   = scale_p[0];
    const float inv = -1.0f / (4.0f * s);

    const bool interior = (gstart >= 0) && (gstart + stage <= n);
    if (interior) {
        // Async global -> LDS, 16 B per lane per instruction.
        // 516 float4s / 256 threads: 2 full rounds + 4 strays.
        #pragma unroll
        for (int k = 0; k < 3; ++k) {
            int v = t + k * NTHREADS;
            if (v < nvec) {
                unsigned lds_off =
                    (unsigned)(size_t)(&smem[v * 4]);          // low 32 bits are the LDS offset
                unsigned long long ga =
                    (unsigned long long)(size_t)(x + gstart + v * 4);
                asm volatile("global_load_async_to_lds_b128 %0, %1, off"
                             :
                             : "v"(lds_off), "v"(ga)
                             : "memory");
            }
        }
        // Each wave waits on its own ASYNCcnt before the block barrier.
        asm volatile("s_wait_asynccnt 0" ::: "memory");
    } else {
        // Boundary blocks (2 of 16384): guarded fill with -inf edge padding.
        for (int i = t; i < stage; i += NTHREADS) {
            int g = gstart + i;
            smem[i] = (g >= 0 && g < n) ? x[g] : -__builtin_inff();
        }
    }
    __syncthreads();

    // h[j] = -(j-5)^2 / (4*scale)
    float h[KSIZE];
    #pragma unroll
    for (int j = 0; j < KSIZE; ++j) {
        float z = (float)(j - KHALF);
        h[j] = z * z * inv;
    }

    // Pull 24 consecutive floats [t*8 .. t*8+23] from LDS as 6x b128.
    float w[3 * VPT];
    const float4* sv = reinterpret_cast<const float4*>(smem);
    #pragma unroll
    for (int c = 0; c < 6; ++c) {
        float4 q = sv[t * 2 + c];
        w[c * 4 + 0] = q.x;
        w[c * 4 + 1] = q.y;
        w[c * 4 + 2] = q.z;
        w[c * 4 + 3] = q.w;
    }

    // out[i0+o] = max_j ( x[i0+o-5+j] + h[j] ); LDS idx of global g is g-base+HALO,
    // so window for output o lives at w[3+o .. 13+o].
    float r[VPT];
    #pragma unroll
    for (int o = 0; o < VPT; ++o) {
        float acc = w[3 + o] + h[0];
        #pragma unroll
        for (int j = 1; j < KSIZE; ++j)
            acc = fmaxf(acc, w[3 + o + j] + h[j]);   // folds to v_max3_num_f32 pairs
        r[o] = acc;
    }

    const int i0 = base + t * VPT;
    if (i0 + VPT <= n) {
        float4* op = reinterpret_cast<float4*>(out + i0);
        op[0] = make_float4(r[0], r[1], r[2], r[3]);
        op[1] = make_float4(r[4], r[5], r[6], r[7]);
    } else {
        #pragma unroll
        for (int o = 0; o < VPT; ++o)
            if (i0 + o < n) out[i0 + o] = r[o];
    }
}

extern "C" void kernel_launch(void* const* d_in, const int* in_sizes, int n_in,
                              void* d_out, int out_size, void* d_ws, size_t ws_size,
                              hipStream_t stream) {
    const float* x     = (const float*)d_in[0];
    const float* scale = (const float*)d_in[1];
    float* out         = (float*)d_out;
    const int n        = in_sizes[0];

    const int nblocks = (n + TILE - 1) / TILE;
    dilation1d_gfx1250_kernel<<<nblocks, NTHREADS, 0, stream>>>(x, scale, out, n);
}